// GNNRecommender_6408091206350
// MI455X (gfx1250) — compile-verified
//
#include <hip/hip_runtime.h>
#include <hip/hip_bf16.h>

#define NUM_USERS 50000
#define NUM_ITEMS 50000
#define N_NODES   100000
#define HIDDEN    128
#define N_EDGES   1600000

typedef __attribute__((ext_vector_type(2))) float v2f;
typedef __attribute__((ext_vector_type(8))) float v8f;

// ---------------------------------------------------------------------------
// Zero a float buffer (n must be a multiple of 4)
// ---------------------------------------------------------------------------
__global__ void zero_f32_kernel(float* __restrict__ p, long n4) {
    long i = (long)blockIdx.x * blockDim.x + threadIdx.x;
    if (i < n4) {
        float4 z = make_float4(0.f, 0.f, 0.f, 0.f);
        ((float4*)p)[i] = z;
    }
}

// ---------------------------------------------------------------------------
// In-degree via f32 atomics: deg[dst] += 1
// ---------------------------------------------------------------------------
__global__ void degree_kernel(const int* __restrict__ edge_index,
                              float* __restrict__ deg) {
    long e = (long)blockIdx.x * blockDim.x + threadIdx.x;
    if (e < N_EDGES) {
        int dst = edge_index[N_EDGES + e];
        atomicAdd(&deg[dst], 1.0f);
    }
}

// deg -> 1/max(deg,1)
__global__ void invdeg_kernel(float* __restrict__ deg) {
    long i = (long)blockIdx.x * blockDim.x + threadIdx.x;
    if (i < N_NODES) {
        deg[i] = 1.0f / fmaxf(deg[i], 1.0f);
    }
}

// ---------------------------------------------------------------------------
// Edge scatter: agg[dst,:] += x[src,:]
// One wave32 per edge, each lane handles 4 consecutive features
// (float4 gather + 4 global f32 atomic adds; all traffic L2-resident).
// ---------------------------------------------------------------------------
__global__ void scatter_kernel(const float* __restrict__ x,
                               const int* __restrict__ edge_index,
                               float* __restrict__ agg) {
    long t    = (long)blockIdx.x * blockDim.x + threadIdx.x;
    long e    = t >> 5;          // edge index
    int  q    = (int)(t & 31);   // lane slot -> features [4q, 4q+3]
    if (e >= N_EDGES) return;
    int src = edge_index[e];
    int dst = edge_index[N_EDGES + e];
    const float4 v = *(const float4*)(x + (long)src * HIDDEN + q * 4);
    float* a = agg + (long)dst * HIDDEN + q * 4;
    atomicAdd(a + 0, v.x);
    atomicAdd(a + 1, v.y);
    atomicAdd(a + 2, v.z);
    atomicAdd(a + 3, v.w);
}

// ---------------------------------------------------------------------------
// Fused layer GEMM:  x = relu( (agg * inv_deg) @ Wl^T + bl + x @ Wr^T )
// Wave-level: each wave owns a 16-row tile x 128 cols (8 accum tiles of 16x16).
// K-loop: 32 steps of v_wmma_f32_16x16x4_f32 against Wl (A = scaled agg),
// then 32 steps against Wr (A = x). In-place store of x (rows independent).
//
// f32 WMMA fragment layouts (ISA 7.12.2):
//   A 16x4 :  lane = (k>=2)*16 + m ; VGPR j holds K = 2*(lane/16)+j
//   B 4x16 :  lane = (k>=2)*16 + n ; VGPR j holds K = 2*(lane/16)+j
//   C/D    :  VGPR r : M = r + 8*(lane/16), N = lane%16
// ---------------------------------------------------------------------------
__global__ void __launch_bounds__(256)
gemm_layer_kernel(const float* __restrict__ agg,
                  const float* __restrict__ inv_deg,
                  float* __restrict__ x,          // in/out
                  const float* __restrict__ Wl,   // [128,128] row-major
                  const float* __restrict__ bl,   // [128]
                  const float* __restrict__ Wr) { // [128,128] row-major
    const int wave = threadIdx.x >> 5;           // 8 waves per block
    const int lane = threadIdx.x & 31;
    const int lr   = lane & 15;                  // m (for A) / n (for B,C)
    const int hi   = lane >> 4;                  // k-half selector

    const long m0 = ((long)blockIdx.x * 8 + wave) * 16;
    if (m0 >= N_NODES) return;

    v8f acc[8];
#pragma unroll
    for (int t = 0; t < 8; ++t) acc[t] = (v8f)(0.0f);

    const long rowA = (m0 + lr) * HIDDEN;        // this lane's A-matrix row
    const float s   = inv_deg[m0 + lr];          // degree norm folded into A

    // ---- phase 1: (agg * inv_deg) @ Wl^T ----
    for (int k0 = 0; k0 < HIDDEN; k0 += 4) {
        const int kk = k0 + 2 * hi;
        v2f a;
        a.x = agg[rowA + kk + 0] * s;
        a.y = agg[rowA + kk + 1] * s;
#pragma unroll
        for (int t = 0; t < 8; ++t) {
            const int n = t * 16 + lr;           // B[k][n] = Wl[n][k]
            v2f b;
            b.x = Wl[n * HIDDEN + kk + 0];
            b.y = Wl[n * HIDDEN + kk + 1];
            acc[t] = __builtin_amdgcn_wmma_f32_16x16x4_f32(
                false, a, false, b, (short)0, acc[t], false, false);
        }
    }

    // ---- phase 2: x @ Wr^T ----
    for (int k0 = 0; k0 < HIDDEN; k0 += 4) {
        const int kk = k0 + 2 * hi;
        v2f a;
        a.x = x[rowA + kk + 0];
        a.y = x[rowA + kk + 1];
#pragma unroll
        for (int t = 0; t < 8; ++t) {
            const int n = t * 16 + lr;
            v2f b;
            b.x = Wr[n * HIDDEN + kk + 0];
            b.y = Wr[n * HIDDEN + kk + 1];
            acc[t] = __builtin_amdgcn_wmma_f32_16x16x4_f32(
                false, a, false, b, (short)0, acc[t], false, false);
        }
    }

    // ---- bias + relu + in-place store ----
#pragma unroll
    for (int t = 0; t < 8; ++t) {
        const int  n    = t * 16 + lr;
        const float bia = bl[n];
#pragma unroll
        for (int r = 0; r < 8; ++r) {
            float v = acc[t][r] + bia;
            v = fmaxf(v, 0.0f);
            x[(m0 + r + 8 * hi) * HIDDEN + n] = v;
        }
    }
}

// ---------------------------------------------------------------------------
// Host-side orchestration
// ---------------------------------------------------------------------------
extern "C" void kernel_launch(void* const* d_in, const int* in_sizes, int n_in,
                              void* d_out, int out_size, void* d_ws, size_t ws_size,
                              hipStream_t stream) {
    const float* user_emb   = (const float*)d_in[0];
    const float* item_emb   = (const float*)d_in[1];
    const float* Wl         = (const float*)d_in[2]; // [2,128,128]
    const float* bl         = (const float*)d_in[3]; // [2,128]
    const float* Wr         = (const float*)d_in[4]; // [2,128,128]
    const int*   edge_index = (const int*)d_in[5];   // [2, N_EDGES]

    float* x = (float*)d_out;                        // [N_NODES,128], final output

    // scratch layout: agg (N_NODES*128 f32) | inv_deg (N_NODES f32)
    float* agg     = (float*)d_ws;
    float* inv_deg = agg + (long)N_NODES * HIDDEN;

    // 1. x = concat(user_emb, item_emb)  (device-to-device, capture-safe)
    hipMemcpyAsync(x, user_emb,
                   (size_t)NUM_USERS * HIDDEN * sizeof(float),
                   hipMemcpyDeviceToDevice, stream);
    hipMemcpyAsync(x + (long)NUM_USERS * HIDDEN, item_emb,
                   (size_t)NUM_ITEMS * HIDDEN * sizeof(float),
                   hipMemcpyDeviceToDevice, stream);

    // 2. inv_deg = 1/max(deg,1)
    {
        long n4 = (long)N_NODES / 4;
        zero_f32_kernel<<<(n4 + 255) / 256, 256, 0, stream>>>(inv_deg, n4);
        degree_kernel<<<(N_EDGES + 255) / 256, 256, 0, stream>>>(edge_index, inv_deg);
        invdeg_kernel<<<(N_NODES + 255) / 256, 256, 0, stream>>>(inv_deg);
    }

    // 3. two GNN layers
    const long aggN4     = (long)N_NODES * HIDDEN / 4;
    const long scatterT  = (long)N_EDGES * 32;
    const int  gemmGrid  = (int)((N_NODES / 16 + 7) / 8);  // 8 row-tiles per block

    for (int l = 0; l < 2; ++l) {
        zero_f32_kernel<<<(aggN4 + 255) / 256, 256, 0, stream>>>(agg, aggN4);
        scatter_kernel<<<(scatterT + 255) / 256, 256, 0, stream>>>(x, edge_index, agg);
        gemm_layer_kernel<<<gemmGrid, 256, 0, stream>>>(
            agg, inv_deg, x,
            Wl + (long)l * HIDDEN * HIDDEN,
            bl + (long)l * HIDDEN,
            Wr + (long)l * HIDDEN * HIDDEN);
    }
}